// Item_Graph_sample_14671608283853
// MI455X (gfx1250) — compile-verified
//
#include <hip/hip_runtime.h>

typedef __attribute__((ext_vector_type(2))) float v2f;
typedef __attribute__((ext_vector_type(4))) float v4f;
typedef __attribute__((ext_vector_type(8))) float v8f;

#define K_NBR 32
#define D_LAT 64

// Opaque uniform-scalar barriers: block InstCombine's select-of-loads ->
// load-of-selected-address fold (round 2 pathology) and the select->index
// fold that produced dynamic-extract cndmask ladders (round 3 pathology).
// Values are wave-uniform, so readfirstlane is semantically a no-op.
static __device__ __forceinline__ int rfl_i(int x) {
  return __builtin_amdgcn_readfirstlane(x);
}
static __device__ __forceinline__ float rfl_f(float x) {
  return __uint_as_float(
      (unsigned)__builtin_amdgcn_readfirstlane((int)__float_as_uint(x)));
}

// One wave32 per item. out[item, :] = sum_k w[k] * features[g[k], :]
//
// V_WMMA_F32_16X16X4_F32, full f32 precision (kernel is L2-bandwidth bound;
// matrix throughput is irrelevant, so no reason to quantize).
//   A (16x4): only row M=0 used = 4 weights of the k-chunk.
//             Lanes 0-15 carry {K0,K1}, lanes 16-31 carry {K2,K3}.
//   B (4x16): column j of accumulator t is feature column 4j+t  (column
//             permutation is legal as long as the store permutes identically),
//             so each lane loads ONE contiguous float4 per gathered row.
//   D row 0 (element 0, lanes 0-15) -> out[item, 4j+t], stored as one b128.
__global__ __launch_bounds__(256)
void item_graph_gather_wmma(const float* __restrict__ features,
                            const int*   __restrict__ item_graph,
                            const float* __restrict__ item_matrix,
                            float* __restrict__ out,
                            int n_items) {
  const int lane = threadIdx.x & 31;
  // wave-uniform item id in an SGPR so weight/index loads scalarize
  const int item = __builtin_amdgcn_readfirstlane(
      (int)(blockIdx.x * (blockDim.x >> 5) + (threadIdx.x >> 5)));
  if (item >= n_items) return;  // uniform; EXEC stays all-1s inside

  const float* __restrict__ wrow = item_matrix + (size_t)item * K_NBR;
  const int*   __restrict__ grow = item_graph  + (size_t)item * K_NBR;

  const int half = lane >> 4;           // 0 -> K pair (0,1), 1 -> K pair (2,3)
  const int j    = lane & 15;           // B column (16-wide)
  const uint32_t jbyte = (uint32_t)(j * 16);  // float4 column offset in a row

  const char* fbase = (const char*)features;

  v8f acc0 = {}, acc1 = {}, acc2 = {}, acc3 = {};

#pragma unroll
  for (int c = 0; c < 8; ++c) {
    const int k0 = 4 * c;
    // uniform scalar reads for this chunk (mergeable into s_load_b128),
    // wrapped in readfirstlane so later selects can't be folded back into
    // the loads
    const float s0 = rfl_f(wrow[k0 + 0]), s1 = rfl_f(wrow[k0 + 1]);
    const float s2 = rfl_f(wrow[k0 + 2]), s3 = rfl_f(wrow[k0 + 3]);
    const int   g0 = rfl_i(grow[k0 + 0]), g1 = rfl_i(grow[k0 + 1]);
    const int   g2 = rfl_i(grow[k0 + 2]), g3 = rfl_i(grow[k0 + 3]);

    // lanes 0-15 need rows (K0,K1); lanes 16-31 need rows (K2,K3)
    const int iEven = half ? g2 : g0;   // v_cndmask between SGPRs
    const int iOdd  = half ? g3 : g1;

    // 32-bit byte offsets against the uniform base -> saddr+voffset loads
    const uint32_t offE = (uint32_t)iEven * (D_LAT * 4u) + jbyte;
    const uint32_t offO = (uint32_t)iOdd  * (D_LAT * 4u) + jbyte;
    const v4f e = *reinterpret_cast<const v4f*>(fbase + offE);  // row K even
    const v4f o = *reinterpret_cast<const v4f*>(fbase + offO);  // row K odd

    // A fragment: row 0 = weights, everything else zero
    const float aE = half ? s2 : s0;
    const float aO = half ? s3 : s1;
    v2f A;
    A[0] = (j == 0) ? aE : 0.0f;
    A[1] = (j == 0) ? aO : 0.0f;

    v2f B0, B1, B2, B3;
    B0[0] = e[0]; B0[1] = o[0];
    B1[0] = e[1]; B1[1] = o[1];
    B2[0] = e[2]; B2[1] = o[2];
    B3[0] = e[3]; B3[1] = o[3];

    acc0 = __builtin_amdgcn_wmma_f32_16x16x4_f32(false, A, false, B0, (short)0, acc0, false, false);
    acc1 = __builtin_amdgcn_wmma_f32_16x16x4_f32(false, A, false, B1, (short)0, acc1, false, false);
    acc2 = __builtin_amdgcn_wmma_f32_16x16x4_f32(false, A, false, B2, (short)0, acc2, false, false);
    acc3 = __builtin_amdgcn_wmma_f32_16x16x4_f32(false, A, false, B3, (short)0, acc3, false, false);
  }

  // D row 0 = element 0 on lanes 0-15; accumulator t holds out[item, 4j+t]
  if (lane < 16) {
    v4f r;
    r[0] = acc0[0]; r[1] = acc1[0]; r[2] = acc2[0]; r[3] = acc3[0];
    char* obase = (char*)out;
    *reinterpret_cast<v4f*>(obase + (uint32_t)item * (D_LAT * 4u) + jbyte) = r;
  }
}

extern "C" void kernel_launch(void* const* d_in, const int* in_sizes, int n_in,
                              void* d_out, int out_size, void* d_ws, size_t ws_size,
                              hipStream_t stream) {
  const float* features    = (const float*)d_in[0];
  const int*   item_graph  = (const int*)d_in[1];   // JAX default x64-off: int32
  const float* item_matrix = (const float*)d_in[2];
  float* out = (float*)d_out;

  const int n_items = in_sizes[2] / K_NBR;  // item_matrix is [N, K]

  const int waves_per_block = 8;            // 256 threads = 8 wave32 = 8 items
  dim3 block(32 * waves_per_block);
  dim3 grid((n_items + waves_per_block - 1) / waves_per_block);
  item_graph_gather_wmma<<<grid, block, 0, stream>>>(features, item_graph,
                                                     item_matrix, out, n_items);
}